// LaneAwareProbabilityLearning_79817672229417
// MI455X (gfx1250) — compile-verified
//
#include <hip/hip_runtime.h>
#include <hip/hip_bf16.h>
#include <math.h>

// ---------------------------------------------------------------------------
// Model dims (compile-time constants from the reference)
// ---------------------------------------------------------------------------
#define INPUT_DIM 3072
#define D_MODEL   128
#define D_INNER   256
#define D_STATE   16
#define D_CONV    4
#define DT_RANK   8
#define NUM_LANES 6
#define BATCH     4
#define SEQLEN    2048
#define NTOK      (BATCH * SEQLEN)   // 8192 tokens

typedef __attribute__((ext_vector_type(2))) float v2f;
typedef __attribute__((ext_vector_type(8))) float v8f;

// ---------------------------------------------------------------------------
// Weight transpose: Bt[n*K + k] = B[k*N + n].  (small, runs once per launch)
// ---------------------------------------------------------------------------
__global__ void transpose_k(const float* __restrict__ B,
                            float* __restrict__ Bt,
                            int K, int N)
{
    const int idx = (int)(blockIdx.x * blockDim.x + threadIdx.x); // K*N
    const int k = idx / N;
    const int n = idx - k * N;
    Bt[(size_t)n * K + k] = B[idx];
}

// ---------------------------------------------------------------------------
// Generic fp32 WMMA GEMM:  C[M,N] = A[M,K] @ B[K,N] (+bias[n]) (+resid[m,n])
// B is supplied TRANSPOSED (Bt[N][K]): each lane's (k,k+1) fragment pair is
// one 8-byte global_load_b64. K and N are template constants so the 8
// B-fragment addresses are ONE base pointer + constant immediate offsets
// (nt*16*K*4 bytes, < 24-bit IOFFSET) -> the compiler can clause the loads
// and count s_wait_loadcnt down across the wmma chain (deep pipelining).
// One wave computes a 16(M) x 128(N) strip: 8 v8f accumulators.
// Launch exactly (M/16)*(N/128) waves; EXEC stays all-ones (WMMA requirement).
//
// WMMA f32 16x16x4 VGPR layouts (wave32):
//   A (16x4):  lane l holds A[l&15][(l>>4)*2 + {0,1}]   -> v2f
//   B (4x16):  lane l holds B[(l>>4)*2 + v][l&15]        -> v2f
//   C/D 16x16: vgpr v, lane l -> row v + (l>>4)*8, col l&15
// ---------------------------------------------------------------------------
template<int K, int N, int NSHIFT>
__global__ void wmma_gemm_f32(const float* __restrict__ A,
                              const float* __restrict__ Bt,     // [N][K]
                              const float* __restrict__ bias,   // may be null
                              const float* __restrict__ resid,  // may be null
                              float* __restrict__ C)
{
    const int gw   = (int)((blockIdx.x * blockDim.x + threadIdx.x) >> 5);
    const int lane = (int)(threadIdx.x & 31);

    const int mt = gw >> NSHIFT;                 // 16-row tile index
    const int nc = gw & ((1 << NSHIFT) - 1);     // 128-col chunk index
    const int m0 = mt * 16;
    const int n0 = nc * 128;

    const int a_row = m0 + (lane & 15);
    const int koff  = (lane >> 4) << 1;          // 0 or 2
    const int ncol  = lane & 15;

    const float* arow = A  + (size_t)a_row * K + koff;
    const float* bbas = Bt + (size_t)(n0 + ncol) * K + koff;  // nt adds 16*K

    v8f acc[8] = {};

#pragma unroll 4
    for (int kk = 0; kk < K; kk += 4) {
        v2f a = *(const v2f*)(arow + kk);                     // global_load_b64
        v2f b[8];
#pragma unroll
        for (int nt = 0; nt < 8; ++nt)                        // 8x b64, const offsets
            b[nt] = *(const v2f*)(bbas + (size_t)nt * 16 * K + kk);
#pragma unroll
        for (int nt = 0; nt < 8; ++nt)
            acc[nt] = __builtin_amdgcn_wmma_f32_16x16x4_f32(
                false, a, false, b[nt], (short)0, acc[nt], false, false);
    }

    const int rbase = m0 + ((lane >> 4) << 3);   // +0 or +8
#pragma unroll
    for (int v = 0; v < 8; ++v) {
        const int m = rbase + v;
#pragma unroll
        for (int nt = 0; nt < 8; ++nt) {
            const int n = n0 + nt * 16 + ncol;
            float val = acc[nt][v];
            if (bias)  val += bias[n];
            if (resid) val += resid[(size_t)m * N + n];
            C[(size_t)m * N + n] = val;
        }
    }
}

// ---------------------------------------------------------------------------
// RMSNorm over D_MODEL=128: one wave per token, 4 elements per lane.
// ---------------------------------------------------------------------------
__global__ void rmsnorm_k(const float* __restrict__ in,
                          const float* __restrict__ w,
                          float* __restrict__ out)
{
    const int tok  = (int)((blockIdx.x * blockDim.x + threadIdx.x) >> 5);
    const int lane = (int)(threadIdx.x & 31);
    const float* row = in + (size_t)tok * D_MODEL;

    float4 v = *(const float4*)(row + lane * 4);
    float s = v.x * v.x + v.y * v.y + v.z * v.z + v.w * v.w;
#pragma unroll
    for (int off = 16; off >= 1; off >>= 1)
        s += __shfl_xor(s, off, 32);
    const float r = rsqrtf(s * (1.0f / D_MODEL) + 1e-5f);

    float* orow = out + (size_t)tok * D_MODEL;
    orow[lane * 4 + 0] = v.x * r * w[lane * 4 + 0];
    orow[lane * 4 + 1] = v.y * r * w[lane * 4 + 1];
    orow[lane * 4 + 2] = v.z * r * w[lane * 4 + 2];
    orow[lane * 4 + 3] = v.w * r * w[lane * 4 + 3];
}

// ---------------------------------------------------------------------------
// Causal depthwise conv (D_CONV=4) over time + SiLU.
// xz layout: (tok, 512); xs = cols [0,256). Output xss: (tok, 256).
// ---------------------------------------------------------------------------
__global__ void conv_silu_k(const float* __restrict__ xz,
                            const float* __restrict__ cw,
                            const float* __restrict__ cb,
                            float* __restrict__ xss)
{
    const size_t idx = (size_t)blockIdx.x * blockDim.x + threadIdx.x; // NTOK*256
    const int c  = (int)(idx & (D_INNER - 1));
    const size_t bt = idx >> 8;               // b*SEQLEN + t
    const int t  = (int)(bt & (SEQLEN - 1));

    float acc = cb[c];
#pragma unroll
    for (int j = 0; j < D_CONV; ++j) {
        const int tt = t - (D_CONV - 1) + j;
        if (tt >= 0)
            acc = fmaf(cw[c * D_CONV + j],
                       xz[(bt - (size_t)(D_CONV - 1 - j)) * (2 * D_INNER) + c],
                       acc);
    }
    xss[idx] = acc / (1.0f + __expf(-acc));   // SiLU
}

// ---------------------------------------------------------------------------
// Per-token: dbl = xss @ W_x (256->40); delta = softplus(dt @ W_dt + b_dt).
// One 256-thread block per token.
// ---------------------------------------------------------------------------
__global__ void xproj_delta_k(const float* __restrict__ xss,
                              const float* __restrict__ Wx,
                              const float* __restrict__ Wdt,
                              const float* __restrict__ bdt,
                              float* __restrict__ Bm,
                              float* __restrict__ Cm,
                              float* __restrict__ delta)
{
    const int tok = (int)blockIdx.x;
    const int tid = (int)threadIdx.x;

    __shared__ float sx[D_INNER];
    __shared__ float sdt[DT_RANK];

    sx[tid] = xss[(size_t)tok * D_INNER + tid];
    __syncthreads();

    if (tid < DT_RANK + 2 * D_STATE) {
        float acc = 0.0f;
        for (int i = 0; i < D_INNER; ++i)
            acc = fmaf(sx[i], Wx[i * (DT_RANK + 2 * D_STATE) + tid], acc);
        if (tid < DT_RANK)                sdt[tid] = acc;
        else if (tid < DT_RANK + D_STATE) Bm[(size_t)tok * D_STATE + (tid - DT_RANK)] = acc;
        else                              Cm[(size_t)tok * D_STATE + (tid - DT_RANK - D_STATE)] = acc;
    }
    __syncthreads();

    float acc = bdt[tid];
#pragma unroll
    for (int r = 0; r < DT_RANK; ++r)
        acc = fmaf(sdt[r], Wdt[r * D_INNER + tid], acc);
    delta[(size_t)tok * D_INNER + tid] = (acc > 20.0f) ? acc : log1pf(__expf(acc));
}

// ---------------------------------------------------------------------------
// Selective scan: h[s] = exp(delta*A[c][s])*h[s] + delta*B[s]*u; y = sum h*C.
// One wave = 2 channels x 16 states; 4x128 = 512 waves total.
// ---------------------------------------------------------------------------
__global__ void scan_k(const float* __restrict__ delta,
                       const float* __restrict__ Bm,
                       const float* __restrict__ Cm,
                       const float* __restrict__ xss,
                       const float* __restrict__ A_log,
                       float* __restrict__ y)
{
    const int gw   = (int)((blockIdx.x * blockDim.x + threadIdx.x) >> 5);  // 0..511
    const int lane = (int)(threadIdx.x & 31);
    const int b = gw >> 7;                                // batch 0..3
    const int c = ((gw & 127) << 1) + (lane >> 4);        // channel 0..255
    const int s = lane & 15;                              // state 0..15

    const float A = -__expf(A_log[c * D_STATE + s]);
    float h = 0.0f;
    const size_t base = (size_t)b * SEQLEN;

    for (int t = 0; t < SEQLEN; ++t) {
        const size_t bt = base + t;
        const float d  = delta[bt * D_INNER + c];
        const float u  = xss[bt * D_INNER + c];
        const float Bv = Bm[bt * D_STATE + s];
        const float Cv = Cm[bt * D_STATE + s];

        const float dA = __expf(d * A);
        h = fmaf(dA, h, d * Bv * u);

        float p = h * Cv;
        p += __shfl_xor(p, 1, 32);
        p += __shfl_xor(p, 2, 32);
        p += __shfl_xor(p, 4, 32);
        p += __shfl_xor(p, 8, 32);
        if (s == 0)
            y[bt * D_INNER + c] = p;
    }
}

// ---------------------------------------------------------------------------
// Gate: y2 = (y + xss*Dp) * silu(res), res = xz[:, 256:512]
// ---------------------------------------------------------------------------
__global__ void gate_k(const float* __restrict__ y,
                       const float* __restrict__ xss,
                       const float* __restrict__ Dp,
                       const float* __restrict__ xz,
                       float* __restrict__ y2)
{
    const size_t idx = (size_t)blockIdx.x * blockDim.x + threadIdx.x;
    const int c  = (int)(idx & (D_INNER - 1));
    const size_t bt = idx >> 8;
    const float r = xz[bt * (2 * D_INNER) + D_INNER + c];
    const float g = r / (1.0f + __expf(-r));
    y2[idx] = (y[idx] + xss[idx] * Dp[c]) * g;
}

// ---------------------------------------------------------------------------
// Final RMSNorm + head (128->6) + softmax + argmax. One wave per token.
// ---------------------------------------------------------------------------
__global__ void head_k(const float* __restrict__ h2,
                       const float* __restrict__ nw,
                       const float* __restrict__ Wh,
                       const float* __restrict__ bh,
                       float* __restrict__ pi,
                       float* __restrict__ preds)
{
    const int tok  = (int)((blockIdx.x * blockDim.x + threadIdx.x) >> 5);
    const int lane = (int)(threadIdx.x & 31);
    const float* row = h2 + (size_t)tok * D_MODEL;

    float4 v = *(const float4*)(row + lane * 4);
    float s = v.x * v.x + v.y * v.y + v.z * v.z + v.w * v.w;
#pragma unroll
    for (int off = 16; off >= 1; off >>= 1)
        s += __shfl_xor(s, off, 32);
    const float r = rsqrtf(s * (1.0f / D_MODEL) + 1e-5f);

    float fm[4];
#pragma unroll
    for (int i = 0; i < 4; ++i)
        fm[i] = ((const float*)&v)[i] * r * nw[lane * 4 + i];

    float lg[NUM_LANES];
#pragma unroll
    for (int j = 0; j < NUM_LANES; ++j) {
        float p = 0.0f;
#pragma unroll
        for (int i = 0; i < 4; ++i)
            p = fmaf(fm[i], Wh[(lane * 4 + i) * NUM_LANES + j], p);
#pragma unroll
        for (int off = 16; off >= 1; off >>= 1)
            p += __shfl_xor(p, off, 32);
        lg[j] = p + bh[j];
    }

    if (lane == 0) {
        float mx = lg[0];
        int best = 0;
#pragma unroll
        for (int j = 1; j < NUM_LANES; ++j)
            if (lg[j] > mx) { mx = lg[j]; best = j; }
        float se = 0.0f;
        float ex[NUM_LANES];
#pragma unroll
        for (int j = 0; j < NUM_LANES; ++j) { ex[j] = __expf(lg[j] - mx); se += ex[j]; }
        const float inv = 1.0f / se;
#pragma unroll
        for (int j = 0; j < NUM_LANES; ++j)
            pi[(size_t)tok * NUM_LANES + j] = ex[j] * inv;
        preds[tok] = (float)best;
    }
}

// ---------------------------------------------------------------------------
// Launcher
// ---------------------------------------------------------------------------
extern "C" void kernel_launch(void* const* d_in, const int* in_sizes, int n_in,
                              void* d_out, int out_size, void* d_ws, size_t ws_size,
                              hipStream_t stream)
{
    const float* x      = (const float*)d_in[0];
    const float* W_emb  = (const float*)d_in[1];
    const float* b_emb  = (const float*)d_in[2];
    const float* norm_w = (const float*)d_in[3];
    const float* W_in   = (const float*)d_in[4];
    const float* conv_w = (const float*)d_in[5];
    const float* conv_b = (const float*)d_in[6];
    const float* W_x    = (const float*)d_in[7];
    const float* W_dt   = (const float*)d_in[8];
    const float* b_dt   = (const float*)d_in[9];
    const float* A_log  = (const float*)d_in[10];
    const float* Dp     = (const float*)d_in[11];
    const float* W_out  = (const float*)d_in[12];
    const float* normf_w= (const float*)d_in[13];
    const float* W_head = (const float*)d_in[14];
    const float* b_head = (const float*)d_in[15];

    float* ws = (float*)d_ws;
    float* h      = ws;                                     // NTOK*128
    float* xn     = h      + (size_t)NTOK * D_MODEL;
    float* xz     = xn     + (size_t)NTOK * D_MODEL;        // NTOK*512
    float* xss    = xz     + (size_t)NTOK * 2 * D_INNER;    // NTOK*256
    float* Bm     = xss    + (size_t)NTOK * D_INNER;        // NTOK*16
    float* Cm     = Bm     + (size_t)NTOK * D_STATE;        // NTOK*16
    float* delta  = Cm     + (size_t)NTOK * D_STATE;        // NTOK*256
    float* yb     = delta  + (size_t)NTOK * D_INNER;        // NTOK*256
    float* y2     = yb     + (size_t)NTOK * D_INNER;        // NTOK*256
    float* h2     = y2     + (size_t)NTOK * D_INNER;        // NTOK*128
    float* Wemb_t = h2     + (size_t)NTOK * D_MODEL;        // 3072*128
    float* Win_t  = Wemb_t + (size_t)INPUT_DIM * D_MODEL;   // 128*512
    float* Wout_t = Win_t  + (size_t)D_MODEL * 2 * D_INNER; // 256*128

    // 0) transpose weights once: Bt[N][K] enables b64 fragment loads
    transpose_k<<<(INPUT_DIM * D_MODEL) / 256, 256, 0, stream>>>(W_emb, Wemb_t, INPUT_DIM, D_MODEL);
    transpose_k<<<(D_MODEL * 2 * D_INNER) / 256, 256, 0, stream>>>(W_in, Win_t, D_MODEL, 2 * D_INNER);
    transpose_k<<<(D_INNER * D_MODEL) / 256, 256, 0, stream>>>(W_out, Wout_t, D_INNER, D_MODEL);

    // 1) h = x @ W_emb + b_emb         (512 waves -> 64 blocks x 256)
    wmma_gemm_f32<INPUT_DIM, D_MODEL, 0><<<64, 256, 0, stream>>>(
        x, Wemb_t, b_emb, nullptr, h);
    // 2) xn = rmsnorm(h, norm_w)
    rmsnorm_k<<<NTOK / 8, 256, 0, stream>>>(h, norm_w, xn);
    // 3) xz = xn @ W_in                (2048 waves -> 256 blocks x 256)
    wmma_gemm_f32<D_MODEL, 2 * D_INNER, 2><<<256, 256, 0, stream>>>(
        xn, Win_t, nullptr, nullptr, xz);
    // 4) xss = silu(conv(xs) + conv_b)
    conv_silu_k<<<NTOK, 256, 0, stream>>>(xz, conv_w, conv_b, xss);
    // 5) dbl projection + delta = softplus(dt @ W_dt + b_dt)
    xproj_delta_k<<<NTOK, 256, 0, stream>>>(xss, W_x, W_dt, b_dt, Bm, Cm, delta);
    // 6) selective scan               (512 waves -> 64 blocks x 256)
    scan_k<<<64, 256, 0, stream>>>(delta, Bm, Cm, xss, A_log, yb);
    // 7) gate
    gate_k<<<NTOK, 256, 0, stream>>>(yb, xss, Dp, xz, y2);
    // 8) h2 = h + y2 @ W_out           (512 waves -> 64 blocks x 256)
    wmma_gemm_f32<D_INNER, D_MODEL, 0><<<64, 256, 0, stream>>>(
        y2, Wout_t, nullptr, h, h2);
    // 9) rmsnorm + head + softmax + argmax
    head_k<<<NTOK / 8, 256, 0, stream>>>(h2, normf_w, W_head, b_head,
                                         (float*)d_out, (float*)d_out + (size_t)NTOK * NUM_LANES);
}